// DistributedMemoryShards_35485019800370
// MI455X (gfx1250) — compile-verified
//
#include <hip/hip_runtime.h>
#include <stdint.h>

// out[b,s,d] = x[b,s,d] * shards[shard_map[d], d]
//
// HBM-bound streaming multiply: 512 MiB moved, 67 MFLOP -> ~22 us floor at
// 23.3 TB/s. Design:
//   1. Per block, stage the 16 KB gathered weight vector w[d] into LDS with
//      CDNA5 async global->LDS loads (per-lane gather, ASYNCcnt), then
//      s_wait_asynccnt 0 + workgroup barrier.
//   2. stride % (dim/4) == 0 (host-enforced) -> each thread's d is invariant,
//      so w4 lives in registers (one hoisted ds_load_b128).
//   3. Uniform scalar trip count (n4/stride identical for all threads) ->
//      loop control is SALU-only; steady state per iteration is exactly
//      global_load_b128 th:NT -> 4x v_mul_f32 -> global_store_b128 th:NT.

typedef float v4f __attribute__((ext_vector_type(4)));

#define MAX_DIM 4096  // reference: DIM = 4096 (16 KB of the 320 KB WGP LDS)

__global__ __launch_bounds__(256)
void shard_scale_kernel(const float* __restrict__ x,
                        const float* __restrict__ shards,
                        const int* __restrict__ smap,
                        float* __restrict__ out,
                        int n4, int dim, int niter) {
  __shared__ float wlds[MAX_DIM];

  // ---- Stage w[d] = shards[smap[d]*dim + d] into LDS via async gather ----
  for (int d = threadIdx.x; d < dim; d += blockDim.x) {
    int s = smap[d];
    uint64_t gaddr = (uint64_t)(uintptr_t)(shards + (size_t)s * (size_t)dim + (size_t)d);
    // Low 32 bits of the flat address of a __shared__ object == wave-relative
    // LDS byte offset, which is what the async op's LDS-address VGPR expects.
    uint32_t ldsOff = (uint32_t)(uintptr_t)&wlds[d];
    asm volatile("global_load_async_to_lds_b32 %0, %1, off"
                 :
                 : "v"(ldsOff), "v"(gaddr)
                 : "memory");
  }
  asm volatile("s_wait_asynccnt 0" ::: "memory");
  __syncthreads();

  // ---- Streaming multiply ----
  const int rowlen4 = dim >> 2;                     // float4 per row (1024)
  const int gtid    = blockIdx.x * blockDim.x + threadIdx.x;
  const int stride  = gridDim.x * blockDim.x;       // host: multiple of rowlen4
  const int d4      = gtid % rowlen4;               // invariant across iters
  const v4f w4      = *(const v4f*)&wlds[d4 << 2];  // one ds_load_b128, hoisted

  const v4f* __restrict__ xv = (const v4f*)x;
  v4f*       __restrict__ ov = (v4f*)out;

  int i = gtid;
#pragma unroll 4
  for (int k = 0; k < niter; ++k, i += stride) {    // uniform count: SALU loop
    v4f v = __builtin_nontemporal_load(xv + i);     // global_load_b128, NT
    v *= w4;
    __builtin_nontemporal_store(v, ov + i);         // global_store_b128, NT
  }
  if (i < n4) {                                     // tail (empty when stride | n4)
    v4f v = __builtin_nontemporal_load(xv + i);
    v *= w4;
    __builtin_nontemporal_store(v, ov + i);
  }
}

static inline long gcdl(long a, long b) {
  while (b) { long t = a % b; a = b; b = t; }
  return a;
}

extern "C" void kernel_launch(void* const* d_in, const int* in_sizes, int n_in,
                              void* d_out, int out_size, void* d_ws, size_t ws_size,
                              hipStream_t stream) {
  const float* x      = (const float*)d_in[0];  // [B, S, DIM] f32
  const float* shards = (const float*)d_in[1];  // [NUM_SHARDS, DIM] f32
  const int*   smap   = (const int*)d_in[2];    // [DIM] int
  float*       out    = (float*)d_out;

  const int dim = in_sizes[2];                  // shard_map length == DIM
  const int n4  = out_size / 4;                 // total float4 elements

  const int threads = 256;                      // 8 wave32 waves per block
  long rowlen4 = dim / 4;
  if (rowlen4 < 1) rowlen4 = 1;
  // blocks must be a multiple of rowlen4/gcd(threads,rowlen4) so that
  // (blocks*threads) % rowlen4 == 0 -> per-thread d is loop-invariant.
  long mult   = rowlen4 / gcdl((long)threads, rowlen4);
  long blocks = 2048;                           // ~32 iters/thread at this size
  blocks = ((blocks + mult - 1) / mult) * mult;

  const long stride = blocks * threads;         // float4 grid stride
  const int  niter  = (int)(n4 / stride);       // uniform full iterations

  shard_scale_kernel<<<(int)blocks, threads, 0, stream>>>(x, shards, smap, out,
                                                          n4, dim, niter);
}